// BasicModel_47871705481510
// MI455X (gfx1250) — compile-verified
//
#include <hip/hip_runtime.h>
#include <hip/hip_bf16.h>

// Per-qubit quantum circuit collapses to:
//   phi       = 0.5f * (w[i] + theta1[i]),  theta1 = {x0, x1, 0, 0, ...}
//   state[i]  = (cos phi, 0, 0, -sin phi)        -> 4 floats at out + 4*i
//   O[i]      = (cos 2phi, 0)                    -> 2 floats at out + 4*n + 2*i
//
// Memory-bound streaming kernel: 4 B in + 24 B out per qubit (~56 MB total,
// ~2.4 us floor at 23.3 TB/s). w is staged through LDS with the CDNA5
// async global->LDS DMA path (ASYNCcnt), read back with ds_load_b128.
// Each lane owns its 16 B LDS slot -> no workgroup barrier needed, only a
// per-wave s_wait_asynccnt / s_wait_dscnt.

#define THREADS      256
#define CHUNK_ELEMS  (THREADS * 4)   // floats of w per block

typedef __attribute__((ext_vector_type(4))) float v4f;

__global__ __launch_bounds__(THREADS) void qubit_rx_kernel(
    const float* __restrict__ x,   // 2 floats
    const float* __restrict__ w,   // n floats
    float* __restrict__ out,       // 6*n floats: [state(4n) | O(2n)]
    int n)
{
    __shared__ __align__(16) float lds[CHUNK_ELEMS];

    const int tid = threadIdx.x;
    const long long base = (long long)blockIdx.x * CHUNK_ELEMS + (long long)tid * 4;

    // LDS byte address of this lane's slot. Low 32 bits of the generic
    // pointer to __shared__ ARE the LDS byte offset (flat-aperture mapping);
    // the ptrtoint also forces the array to escape
    // so the asm "memory" clobbers are honored for it.
    float* lslot = &lds[tid * 4];
    const unsigned lds_off = (unsigned)(uintptr_t)lslot;

    // ---- CDNA5 async copy: 16 B of w per lane, global -> LDS ----
    if (base < n) {
        const float* gaddr = w + base;   // 64-bit addr -> VGPR pair (GV mode)
        asm volatile("global_load_async_to_lds_b128 %0, %1, off"
                     :: "v"(lds_off), "v"(gaddr)
                     : "memory");
    }
    // wait for this wave's async transfers (ASYNCcnt -> 0)
    asm volatile("s_wait_asynccnt 0x0" ::: "memory");

    if (base >= n) return;

    // explicit LDS read-back of this lane's own slot (cannot be folded)
    v4f wv;
    asm volatile("ds_load_b128 %0, %1\n\t"
                 "s_wait_dscnt 0x0"
                 : "=v"(wv)
                 : "v"(lds_off)
                 : "memory");

    float t[4] = { wv.x, wv.y, wv.z, wv.w };
    if (base == 0) {            // qubits 0,1 also get the data-encoding Rx(x)
        t[0] += x[0];
        t[1] += x[1];
    }

    float c[4], s[4];
    float4 st[4];
    #pragma unroll
    for (int k = 0; k < 4; ++k) {
        float phi = 0.5f * t[k];
        c[k] = __cosf(phi);     // v_cos_f32
        s[k] = __sinf(phi);     // v_sin_f32
        // state row: [re(s0), im(s0), re(s1), im(s1)]
        st[k] = make_float4(c[k], 0.0f, 0.0f, -s[k]);
    }

    // O = cos^2 - sin^2 = cos(2*phi), imag exactly 0
    float4 o0 = make_float4(c[0]*c[0] - s[0]*s[0], 0.0f,
                            c[1]*c[1] - s[1]*s[1], 0.0f);
    float4 o1 = make_float4(c[2]*c[2] - s[2]*s[2], 0.0f,
                            c[3]*c[3] - s[3]*s[3], 0.0f);

    // state: 16 contiguous floats per lane, 16B-aligned
    float4* outs = reinterpret_cast<float4*>(out + 4 * base);
    #pragma unroll
    for (int k = 0; k < 4; ++k) outs[k] = st[k];

    // O: 8 contiguous floats per lane at offset 4*n
    float4* outo = reinterpret_cast<float4*>(out + 4LL * n + 2 * base);
    outo[0] = o0;
    outo[1] = o1;
}

extern "C" void kernel_launch(void* const* d_in, const int* in_sizes, int n_in,
                              void* d_out, int out_size, void* d_ws, size_t ws_size,
                              hipStream_t stream) {
    const float* x = (const float*)d_in[0];   // (2,)
    const float* w = (const float*)d_in[1];   // (N_QUBITS,)
    float* out = (float*)d_out;               // 6*N_QUBITS floats
    const int n = in_sizes[1];

    const int blocks = (n + CHUNK_ELEMS - 1) / CHUNK_ELEMS;
    qubit_rx_kernel<<<blocks, THREADS, 0, stream>>>(x, w, out, n);
}